// CapsuleLayer_51488067944654
// MI455X (gfx1250) — compile-verified
//
#include <hip/hip_runtime.h>
#include <hip/hip_bf16.h>
#include <math.h>

// ---------------------------------------------------------------------------
// CapsuleLayer dynamic routing, fused for MI455X (gfx1250, wave32).
//
// u_hat (189 MB) is never materialized; each routing pass recomputes it from
// L2-resident W (5.9 MB) and x (9.4 MB) with V_WMMA_F32_16X16X4_F32
// (M = co-tile, N = batch-tile, K = i). Softmax normalization is deferred
// (unnormalized exp + Z accumulator) so each iteration costs one W pass.
//
// K-permutation trick: the i-contraction is order-invariant, so the two K=4
// groups use i-sets {0,1,4,5} and {2,3,6,7}. With the ISA's A/B operand
// layout (lane-half h holds K = {2h, 2h+1}) every lane then needs a
// contiguous i-range {4h .. 4h+3}: A = one global_load_b128 per (r, tile),
// B = one ds_load_b128 per r.
// ---------------------------------------------------------------------------

typedef __attribute__((ext_vector_type(2))) float v2f;
typedef __attribute__((ext_vector_type(4))) float v4f;
typedef __attribute__((ext_vector_type(8))) float v8f;

namespace caps {
constexpr int B = 256, R = 1152, C = 10, O = 16, I = 8;
constexpr int CO = C * O;              // 160
constexpr int BTILE = 16;              // batch tile  (WMMA N)
constexpr int RCHUNK = 128;            // r per block
constexpr int NTHREADS = 256;          // 8 wave32s
constexpr int RPW = RCHUNK / (NTHREADS / 32); // 16 r per wave
}

__device__ __forceinline__ v8f wmma4(v2f a, v2f b, v8f c) {
  // D = A(16x4 f32) x B(4x16 f32) + C ; exact fp32 path
  return __builtin_amdgcn_wmma_f32_16x16x4_f32(
      /*neg_a=*/false, a, /*neg_b=*/false, b,
      /*c_mod=*/(short)0, c, /*reuse_a=*/false, /*reuse_b=*/false);
}

// MODE 0: iter0, e == 1 (uniform softmax), Z fixed = R. Accumulate u_hat sums.
// MODE 1: iter1, b_new = u_hat . v0, write b to ws, accumulate e*u_hat and e.
// MODE 2: iter2, b_new = b_prev + u_hat . v1, accumulate e*u_hat and e.
template <int MODE>
__global__ __launch_bounds__(caps::NTHREADS)
void caps_pass(const float* __restrict__ x, const float* __restrict__ W,
               float* __restrict__ tj, float* __restrict__ Zacc,
               const float* __restrict__ vprev, float* __restrict__ bws) {
  using namespace caps;
  __shared__ float xs[BTILE * RCHUNK * I];           // 64 KB x tile
  __shared__ float vs[MODE ? (BTILE * CO) : 4];      // 10 KB v tile (modes 1,2)

  const int tid = threadIdx.x;
  const int b0 = blockIdx.x * BTILE;
  const int r0 = blockIdx.y * RCHUNK;

  // ---- stage x[b0:b0+16, r0:r0+128, :] into LDS as [bl][rl][i] -------------
  #pragma unroll
  for (int k = 0; k < 16; ++k) {
    int q = tid + k * NTHREADS;          // float4 index, 4096 total
    int bl = q >> 8;                     // 256 float4 per b row-block
    int rl = (q >> 1) & (RCHUNK - 1);
    int ih = (q & 1) * 4;
    float4 val = *(const float4*)&x[((size_t)(b0 + bl) * R + (r0 + rl)) * I + ih];
    *(float4*)&xs[(size_t)q * 4] = val;
  }
  if (MODE) {  // stage v[b0:b0+16, :, :] as [bl][c*16+o]
    for (int q = tid; q < (BTILE * CO) / 4; q += NTHREADS) {
      int bl = q / 40;
      int rem = (q % 40) * 4;
      float4 val = *(const float4*)&vprev[(size_t)(b0 + bl) * CO + rem];
      *(float4*)&vs[(size_t)q * 4] = val;
    }
  }
  __syncthreads();

  const int wave = tid >> 5;
  const int lane = tid & 31;
  const int half = lane >> 4;   // selects K pair / D's M-high half
  const int l16 = lane & 15;    // A: M row ; B/D: N (= batch lane)

  const v8f zero8 = {0.f, 0.f, 0.f, 0.f, 0.f, 0.f, 0.f, 0.f};
  v8f acc[C];
  float zpart[C];
  #pragma unroll
  for (int t = 0; t < C; ++t) { acc[t] = zero8; zpart[t] = 0.f; }

  for (int j = 0; j < RPW; ++j) {
    const int rl = wave * RPW + j;
    const int rg = r0 + rl;
    const float* __restrict__ wrow = W + (size_t)rg * CO * I;
    if (j + 1 < RPW) __builtin_prefetch(wrow + CO * I, 0, 1);  // next r's W slice

    // B operand = x^T tile: one b128 per lane covers both K-groups
    // (K-group 0 -> i {0,1}/{4,5}, K-group 1 -> i {2,3}/{6,7} per lane-half)
    const float* xr = &xs[((size_t)l16 * RCHUNK + rl) * I];
    v4f x4 = *(const v4f*)&xr[4 * half];
    v2f bop0 = __builtin_shufflevector(x4, x4, 0, 1);
    v2f bop1 = __builtin_shufflevector(x4, x4, 2, 3);

    #pragma unroll
    for (int t = 0; t < C; ++t) {   // c == t, o runs over the 16-row M tile
      // A operand = W[r, t*16+M, i]: one b128 per lane, same K-permutation
      const float* wl = wrow + ((t * 16 + l16) * I + 4 * half);
      v4f w4 = *(const v4f*)&wl[0];
      v2f aop0 = __builtin_shufflevector(w4, w4, 0, 1);
      v2f aop1 = __builtin_shufflevector(w4, w4, 2, 3);

      if (MODE == 0) {
        // e == 1: accumulate u_hat sums in place through the C operand
        acc[t] = wmma4(aop0, bop0, acc[t]);
        acc[t] = wmma4(aop1, bop1, acc[t]);
      } else {
        v8f u = wmma4(aop0, bop0, zero8);   // u_hat[o,b] tile
        u = wmma4(aop1, bop1, u);
        // b_new contribution: sum_o u[o,b] * v[b, c=t, o]
        const float* vrw = &vs[(size_t)l16 * CO + t * 16 + 8 * half];
        v4f vlo = *(const v4f*)&vrw[0];
        v4f vhi = *(const v4f*)&vrw[4];
        float p = u[0] * vlo.x + u[1] * vlo.y + u[2] * vlo.z + u[3] * vlo.w +
                  u[4] * vhi.x + u[5] * vhi.y + u[6] * vhi.z + u[7] * vhi.w;
        p += __shfl_xor(p, 16, 32);  // combine o<8 and o>=8 halves
        float bnew = p;
        const size_t bidx = ((size_t)rg * C + t) * B + b0 + l16;
        if (MODE == 2) bnew += bws[bidx];
        if (MODE == 1) { if (half == 0) bws[bidx] = bnew; }
        float e = __expf(bnew);      // unnormalized softmax weight
        zpart[t] += e;
        #pragma unroll
        for (int v = 0; v < 8; ++v) acc[t][v] += e * u[v];
      }
    }
  }

  // ---- cross-block reduction of s_j numerator (and Z denominator) ----------
  #pragma unroll
  for (int t = 0; t < C; ++t) {
    #pragma unroll
    for (int v = 0; v < 8; ++v) {
      int o = v + 8 * half;
      atomicAdd(&tj[(size_t)(t * O + o) * B + b0 + l16], acc[t][v]);
    }
    if (MODE && half == 0)
      atomicAdd(&Zacc[(size_t)t * B + b0 + l16], zpart[t]);
  }
}

// squash: v = (|s|^2/(1+|s|^2)) * s/sqrt(|s|^2+eps), s = t/Z. One thread per (b,c).
__global__ __launch_bounds__(caps::NTHREADS)
void caps_squash(float* __restrict__ tj, float* __restrict__ Zacc,
                 float* __restrict__ vout, float fixedZ, int clear) {
  using namespace caps;
  int idx = blockIdx.x * blockDim.x + threadIdx.x;
  if (idx >= B * C) return;
  int b = idx / C, c = idx % C;
  float z = (fixedZ > 0.f) ? fixedZ : Zacc[(size_t)c * B + b];
  float inv = 1.f / z;
  float s[O];
  float sq = 0.f;
  #pragma unroll
  for (int o = 0; o < O; ++o) {
    s[o] = tj[(size_t)(c * O + o) * B + b] * inv;
    sq += s[o] * s[o];
  }
  float scale = (sq / (1.f + sq)) / sqrtf(sq + 1e-7f);
  #pragma unroll
  for (int o = 0; o < O; ++o)
    vout[(size_t)b * CO + c * O + o] = scale * s[o];
  if (clear) {  // re-zero accumulators for the next routing pass
    #pragma unroll
    for (int o = 0; o < O; ++o) tj[(size_t)(c * O + o) * B + b] = 0.f;
    Zacc[(size_t)c * B + b] = 0.f;
  }
}

extern "C" void kernel_launch(void* const* d_in, const int* in_sizes, int n_in,
                              void* d_out, int out_size, void* d_ws, size_t ws_size,
                              hipStream_t stream) {
  using namespace caps;
  (void)in_sizes; (void)n_in; (void)out_size; (void)ws_size;
  const float* x = (const float*)d_in[0];   // (256,1152,8) f32
  const float* W = (const float*)d_in[1];   // (1152,10,16,8) f32
  float* out = (float*)d_out;               // (256,1,10,16,1) f32 == [b][c][o]

  float* ws = (float*)d_ws;                 // ws layout (floats):
  float* tj  = ws;                          // [0, 40960)      s_j numerators [co][b]
  float* Z   = ws + (size_t)B * C * O;      // [40960, 43520)  softmax denoms [c][b]
  float* vws = Z + (size_t)B * C;           // [43520, 84480)  v_k            [b][c][o]
  float* bws = vws + (size_t)B * C * O;     // [84480, +2.95M) b_ij           [r][c][b]

  hipMemsetAsync(tj, 0, (size_t)(B * C * O + B * C) * sizeof(float), stream);

  dim3 grid(B / BTILE, R / RCHUNK);         // (16, 9)
  dim3 blk(NTHREADS);
  dim3 sgrid((B * C + NTHREADS - 1) / NTHREADS);

  // iter 0: uniform softmax (Z = R analytically)
  caps_pass<0><<<grid, blk, 0, stream>>>(x, W, tj, Z, nullptr, nullptr);
  caps_squash<<<sgrid, blk, 0, stream>>>(tj, Z, vws, (float)R, 1);
  // iter 1: b1 = u_hat . v0
  caps_pass<1><<<grid, blk, 0, stream>>>(x, W, tj, Z, vws, bws);
  caps_squash<<<sgrid, blk, 0, stream>>>(tj, Z, vws, 0.f, 1);
  // iter 2 (last): b2 = b1 + u_hat . v1 ; write final v to d_out
  caps_pass<2><<<grid, blk, 0, stream>>>(x, W, tj, Z, vws, bws);
  caps_squash<<<sgrid, blk, 0, stream>>>(tj, Z, out, 0.f, 0);
}